// NeuralUniLasso_26061861552577
// MI455X (gfx1250) — compile-verified
//
#include <hip/hip_runtime.h>

// Problem constants (match reference): B=8192, F=1024, H=8
#define FDIM 1024
#define HDIM 8

typedef __attribute__((ext_vector_type(2))) float v2f;
typedef __attribute__((ext_vector_type(8))) float v8f;

// -------------------------------------------------------------------------
// Kernel 1: streaming Z pass.  block = (32 features, 8 rows), each thread
// owns ONE feature (25 weight floats register-cached from LDS) and loops
// over 8 rows -> block covers 32 features x 64 rows.  x loads / Z stores
// are lane-coalesced along f.  Weights staged to LDS once per block.
// grid = (F/32, B/64)
// -------------------------------------------------------------------------
__global__ __launch_bounds__(256) void nul_z_kernel(
    const float* __restrict__ x,  const float* __restrict__ W1,
    const float* __restrict__ b1, const float* __restrict__ W2,
    const float* __restrict__ b2, float* __restrict__ Z)
{
    __shared__ float sW1[HDIM][32];
    __shared__ float sB1[HDIM][32];
    __shared__ float sW2[HDIM][32];
    __shared__ float sB2[32];

    const int ftile = blockIdx.x * 32;
    const int tid   = threadIdx.y * 32 + threadIdx.x;

    // Cooperative staged load: W1/b1/W2 are [F][H] row-major.
    if (tid < 32 * HDIM) {                 // 256 threads load 256 elems each
        const int f = tid >> 3, h = tid & 7;
        const int g = (ftile + f) * HDIM + h;
        sW1[h][f] = W1[g];
        sB1[h][f] = b1[g];
        sW2[h][f] = W2[g];
        if (h == 0) sB2[f] = b2[ftile + f];
    }
    __syncthreads();

    // Register-cache this thread's single feature.
    const int fl = threadIdx.x;
    float w1r[HDIM], b1r[HDIM], w2r[HDIM];
    #pragma unroll
    for (int h = 0; h < HDIM; ++h) {
        w1r[h] = sW1[h][fl];
        b1r[h] = sB1[h][fl];
        w2r[h] = sW2[h][fl];
    }
    const float b2r = sB2[fl];

    const int fg   = ftile + fl;
    const int row0 = blockIdx.y * 64 + threadIdx.y;

    #pragma unroll
    for (int r = 0; r < 8; ++r) {
        const int row = row0 + r * 8;
        const float xv = x[(size_t)row * FDIM + fg];
        float acc = b2r;
        #pragma unroll
        for (int h = 0; h < HDIM; ++h) {
            float t = fmaf(xv, w1r[h], b1r[h]);
            t = fmaxf(t, 0.0f);
            acc = fmaf(t, w2r[h], acc);
        }
        Z[(size_t)row * FDIM + fg] = acc;
    }
}

// -------------------------------------------------------------------------
// Kernel 2: positive_weights = relu(fusion_weights)   (F elements)
// -------------------------------------------------------------------------
__global__ void nul_pw_kernel(const float* __restrict__ fw,
                              float* __restrict__ pw, int n)
{
    const int i = blockIdx.x * blockDim.x + threadIdx.x;
    if (i < n) pw[i] = fmaxf(fw[i], 0.0f);
}

// -------------------------------------------------------------------------
// Kernel 3: y = Z @ pw + bias via V_WMMA_F32_16X16X4_F32.
// One wave32 per 16-row tile of B.  Per WMMA step (K=4 features):
//   A (16x4 f32, 2 VGPR):  lane L holds rows m=L&15, cols {2*(L>>4), +1}
//   B (4x16 f32, 2 VGPR):  pw[f0+k] replicated across all N columns
//     (B layout K-index expression == A layout -> same lane formula)
// => D[m][n] = sum_k Z[m,k]*pw[k] for every n.  Four independent K-chunk
// accumulators for ILP, elementwise-summed at the end.  Lanes 0/16 hold
// the N=0 column for M=0..7 / 8..15 and write out 8 rows each.
// block = 256 threads = 8 waves; grid = B / (8*16)
// -------------------------------------------------------------------------
__global__ __launch_bounds__(256) void nul_y_wmma_kernel(
    const float* __restrict__ Z, const float* __restrict__ pw,
    const float* __restrict__ bias, float* __restrict__ y)
{
    const int lane  = threadIdx.x & 31;
    const int wave  = threadIdx.x >> 5;
    const int tile  = (blockIdx.x * 8 + wave) * 16;      // row base
    const int m     = lane & 15;
    const int khalf = lane >> 4;                          // 0 or 1

    const float* zrow = Z + (size_t)(tile + m) * FDIM + 2 * khalf;
    const float* pwp  = pw + 2 * khalf;

    v8f c0 = {}, c1 = {}, c2 = {}, c3 = {};
    for (int f0 = 0; f0 < FDIM; f0 += 16) {
        v2f a0 = *(const v2f*)(zrow + f0);
        v2f b0 = *(const v2f*)(pwp  + f0);
        v2f a1 = *(const v2f*)(zrow + f0 + 4);
        v2f b1 = *(const v2f*)(pwp  + f0 + 4);
        v2f a2 = *(const v2f*)(zrow + f0 + 8);
        v2f b2 = *(const v2f*)(pwp  + f0 + 8);
        v2f a3 = *(const v2f*)(zrow + f0 + 12);
        v2f b3 = *(const v2f*)(pwp  + f0 + 12);
        c0 = __builtin_amdgcn_wmma_f32_16x16x4_f32(false, a0, false, b0,
                                                   (short)0, c0, false, false);
        c1 = __builtin_amdgcn_wmma_f32_16x16x4_f32(false, a1, false, b1,
                                                   (short)0, c1, false, false);
        c2 = __builtin_amdgcn_wmma_f32_16x16x4_f32(false, a2, false, b2,
                                                   (short)0, c2, false, false);
        c3 = __builtin_amdgcn_wmma_f32_16x16x4_f32(false, a3, false, b3,
                                                   (short)0, c3, false, false);
    }
    v8f c = (c0 + c1) + (c2 + c3);

    const float bs = bias[0];
    if (m == 0) {                       // lanes 0 (M=0..7) and 16 (M=8..15)
        float* yp = y + tile + 8 * khalf;
        #pragma unroll
        for (int r = 0; r < 8; ++r) yp[r] = c[r] + bs;
    }
}

// -------------------------------------------------------------------------
// Launch: d_in order = x, W1, b1, W2, b2, fusion_weights, bias
// d_out  = [ y_pred (B) | Z (B*F) | positive_weights (F) ]
// -------------------------------------------------------------------------
extern "C" void kernel_launch(void* const* d_in, const int* in_sizes, int n_in,
                              void* d_out, int out_size, void* d_ws, size_t ws_size,
                              hipStream_t stream)
{
    (void)n_in; (void)out_size; (void)d_ws; (void)ws_size;

    const float* x    = (const float*)d_in[0];
    const float* W1   = (const float*)d_in[1];
    const float* b1   = (const float*)d_in[2];
    const float* W2   = (const float*)d_in[3];
    const float* b2   = (const float*)d_in[4];
    const float* fw   = (const float*)d_in[5];
    const float* bias = (const float*)d_in[6];

    const int F = in_sizes[4];                 // 1024
    const int B = in_sizes[0] / F;             // 8192

    float* y  = (float*)d_out;                 // [B]
    float* Z  = y + B;                         // [B*F]
    float* pw = Z + (size_t)B * F;             // [F]

    // 1) streaming Z pass
    dim3 zblk(32, 8);
    dim3 zgrd(F / 32, B / 64);
    nul_z_kernel<<<zgrd, zblk, 0, stream>>>(x, W1, b1, W2, b2, Z);

    // 2) positive weights (kernel 3 reads them from d_out, same stream)
    nul_pw_kernel<<<(F + 255) / 256, 256, 0, stream>>>(fw, pw, F);

    // 3) y = Z @ pw + bias via WMMA (Z re-read hits L2: 33.5 MB << 192 MB)
    nul_y_wmma_kernel<<<B / 128, 256, 0, stream>>>(Z, pw, bias, y);
}